// decoder_attn_49933289783868
// MI455X (gfx1250) — compile-verified
//
#include <hip/hip_runtime.h>
#include <math.h>

typedef __attribute__((ext_vector_type(2))) float v2f;
typedef __attribute__((ext_vector_type(8))) float v8f;

#define BB 64
#define SRC 128
#define EE 100
#define HH 1000
#define HD 500
#define VV 32000
#define KB 100   // K-chunk staged in LDS; all K used (100/500/1000/2000) are multiples

// ---------------- embedding gather (row 0 forced to zero) ----------------
__global__ void k_embed(const int* __restrict__ x, const float* __restrict__ embed,
                        float* __restrict__ e) {
  int i = blockIdx.x * blockDim.x + threadIdx.x;
  if (i >= BB * EE) return;
  int b = i / EE, j = i % EE;
  int idx = x[b];
  e[i] = (idx == 0) ? 0.0f : embed[(size_t)idx * EE + j];
}

// ---------------- generic M=64 WMMA f32 GEMM, LDS-staged A ----------------
// D[64 x N] = A[64 x K] (row-major, stride lda) @ B + bias, optional tanh.
// BT=true : B given as W[N x K] row-major  (D = A @ W^T)
// BT=false: B given as W[K x N] row-major  (D = A @ W)
// Block = 8 waves; each wave owns a 64x16 tile (4 M-tiles reuse each B fragment).
// A chunk (64 x KB) is staged in LDS once per block and shared by all 8 waves;
// B (the HBM-bound stream) is software-pipelined one k-step ahead.
template<bool BT>
__global__ void k_gemm64(const float* __restrict__ A, int lda,
                         const float* __restrict__ Wt,
                         const float* __restrict__ bias,
                         float* __restrict__ D, int ldd,
                         int N, int K, int act)
{
  __shared__ float sA[BB * KB];                     // 25.6 KB
  const int tid  = threadIdx.x;
  const int lane = tid & 31;
  const int wave = tid >> 5;
  const int tile = blockIdx.x * (blockDim.x >> 5) + wave;
  const int n0   = tile * 16;
  const bool active = (n0 < N);                     // wave-uniform
  const int half = lane >> 4;                       // 0: K lanes {0,1}; 1: {2,3}
  const int l16  = lane & 15;
  int colB = n0 + l16;
  if (colB >= N) colB = N - 1;                      // clamp loads on ragged tile

  const float* bpT = Wt + (size_t)colB * K;         // BT row pointer

  // LDS fragment addresses for the 4 M-tiles (conflict-free: 100-dword row stride)
  const float* sa0 = sA + (size_t)(l16 +  0) * KB;
  const float* sa1 = sA + (size_t)(l16 + 16) * KB;
  const float* sa2 = sA + (size_t)(l16 + 32) * KB;
  const float* sa3 = sA + (size_t)(l16 + 48) * KB;

  v8f c0 = {}, c1 = {}, c2 = {}, c3 = {};

  for (int kc = 0; kc < K; kc += KB) {
    __syncthreads();                                // protect previous chunk reads
    // cooperative A-chunk load: 64*KB / 256 = 25 elems/thread, contiguous per row
    for (int idx = tid; idx < BB * KB; idx += 256) {
      int r = idx / KB, c = idx % KB;
      sA[idx] = A[(size_t)r * lda + kc + c];
    }
    __syncthreads();

    if (active) {                                   // wave-uniform: EXEC all-ones
      // prime the B pipeline for this chunk
      v2f bcur;
      if (BT) {
        bcur = *(const v2f*)(bpT + kc + 2 * half);
      } else {
        bcur.x = Wt[(size_t)(kc + 2 * half)     * N + colB];
        bcur.y = Wt[(size_t)(kc + 2 * half + 1) * N + colB];
      }
      for (int k = 0; k < KB; k += 4) {
        const int kk = k + 2 * half;
        // prefetch next B fragment (one global load in flight behind the WMMAs)
        v2f bnext = bcur;
        if (k + 4 < KB) {
          if (BT) {
            bnext = *(const v2f*)(bpT + kc + kk + 4);
          } else {
            bnext.x = Wt[(size_t)(kc + kk + 4) * N + colB];
            bnext.y = Wt[(size_t)(kc + kk + 5) * N + colB];
          }
        }
        v2f af;
        af = *(const v2f*)(sa0 + kk);
        c0 = __builtin_amdgcn_wmma_f32_16x16x4_f32(false, af, false, bcur, (short)0, c0, false, false);
        af = *(const v2f*)(sa1 + kk);
        c1 = __builtin_amdgcn_wmma_f32_16x16x4_f32(false, af, false, bcur, (short)0, c1, false, false);
        af = *(const v2f*)(sa2 + kk);
        c2 = __builtin_amdgcn_wmma_f32_16x16x4_f32(false, af, false, bcur, (short)0, c2, false, false);
        af = *(const v2f*)(sa3 + kk);
        c3 = __builtin_amdgcn_wmma_f32_16x16x4_f32(false, af, false, bcur, (short)0, c3, false, false);
        bcur = bnext;
      }
    }
  }

  const int col = n0 + l16;
  if (!active || col >= N) return;                  // after all barriers: safe
  const float bv = bias ? bias[col] : 0.0f;
  const int mofs = half * 8;                        // C/D layout: lanes>=16 hold M+8
  v8f accs[4] = {c0, c1, c2, c3};
#pragma unroll
  for (int i = 0; i < 4; ++i) {
#pragma unroll
    for (int r = 0; r < 8; ++r) {
      float v = accs[i][r] + bv;
      if (act) v = tanhf(v);
      D[(size_t)(i * 16 + mofs + r) * ldd + col] = v;
    }
  }
}

// ---------------- GRU gate fusion ----------------
__global__ void k_gate(const float* __restrict__ gi, const float* __restrict__ gh,
                       const float* __restrict__ hprev,
                       float* __restrict__ out, int out_off, int out_ld)
{
  int i = blockIdx.x * blockDim.x + threadIdx.x;
  if (i >= BB * HD) return;
  int b = i / HD, j = i % HD;
  const float* gib = gi + (size_t)b * 3 * HD;
  const float* ghb = gh + (size_t)b * 3 * HD;
  float ir = gib[j], iz = gib[HD + j], in = gib[2 * HD + j];
  float hr = ghb[j], hz = ghb[HD + j], hn = ghb[2 * HD + j];
  float r = 1.0f / (1.0f + expf(-(ir + hr)));
  float z = 1.0f / (1.0f + expf(-(iz + hz)));
  float n = tanhf(in + r * hn);
  float h = hprev[(size_t)b * HD + j];
  out[(size_t)b * out_ld + out_off + j] = (1.0f - z) * n + z * h;
}

// ---------------- wave reductions (wave32) ----------------
__device__ __forceinline__ float waveSum(float v) {
  v += __shfl_xor(v, 16, 32); v += __shfl_xor(v, 8, 32);
  v += __shfl_xor(v, 4, 32);  v += __shfl_xor(v, 2, 32);
  v += __shfl_xor(v, 1, 32);
  return v;
}
__device__ __forceinline__ float waveMax(float v) {
  v = fmaxf(v, __shfl_xor(v, 16, 32)); v = fmaxf(v, __shfl_xor(v, 8, 32));
  v = fmaxf(v, __shfl_xor(v, 4, 32));  v = fmaxf(v, __shfl_xor(v, 2, 32));
  v = fmaxf(v, __shfl_xor(v, 1, 32));
  return v;
}

// ---------------- attention: scores -> softmax -> context ----------------
// scores[b,t] = dot(hW[b], enc[b,t]) (h.b_fnn term is softmax-invariant, dropped)
__global__ void k_attention(float* __restrict__ hcat, const float* __restrict__ hW,
                            const float* __restrict__ enc)
{
  __shared__ float sW[HH];
  __shared__ float sS[SRC];
  __shared__ float sRed[8];
  __shared__ float sMax, sSum;
  const int b = blockIdx.x;
  const int tid = threadIdx.x;
  const int lane = tid & 31, wid = tid >> 5;

  for (int i = tid; i < HH; i += blockDim.x) sW[i] = hW[(size_t)b * HH + i];
  __syncthreads();

  const float* encb = enc + (size_t)b * SRC * HH;
  for (int t = wid; t < SRC; t += 8) {
    const float* row = encb + (size_t)t * HH;
    float p = 0.0f;
    for (int e = lane; e < HH; e += 32) p += sW[e] * row[e];
    p = waveSum(p);
    if (lane == 0) sS[t] = p;
  }
  __syncthreads();

  float v = (tid < SRC) ? sS[tid] : -INFINITY;
  float m = waveMax(v);
  if (lane == 0) sRed[wid] = m;
  __syncthreads();
  if (tid == 0) {
    float mm = sRed[0];
    for (int i = 1; i < 8; ++i) mm = fmaxf(mm, sRed[i]);
    sMax = mm;
  }
  __syncthreads();
  float ex = (tid < SRC) ? expf(v - sMax) : 0.0f;
  float s = waveSum(ex);
  if (lane == 0) sRed[wid] = s;
  __syncthreads();
  if (tid == 0) {
    float ss = 0.0f;
    for (int i = 0; i < 8; ++i) ss += sRed[i];
    sSum = ss;
  }
  __syncthreads();
  if (tid < SRC) sS[tid] = ex / sSum;
  __syncthreads();

  for (int d = tid; d < HH; d += blockDim.x) {
    float acc = 0.0f;
    for (int t = 0; t < SRC; ++t) acc += sS[t] * encb[(size_t)t * HH + d];
    hcat[(size_t)b * 2 * HH + HH + d] = acc;
  }
}

// ---------------- per-row logsumexp over V=32000 ----------------
__global__ void k_lse(const float* __restrict__ logits, float* __restrict__ lse)
{
  __shared__ float sRed[8];
  __shared__ float sM;
  const int b = blockIdx.x, tid = threadIdx.x;
  const int lane = tid & 31, wid = tid >> 5;
  const float* row = logits + (size_t)b * VV;
  float m = -INFINITY;
  for (int i = tid; i < VV; i += blockDim.x) m = fmaxf(m, row[i]);
  m = waveMax(m);
  if (lane == 0) sRed[wid] = m;
  __syncthreads();
  if (tid == 0) {
    float mm = sRed[0];
    for (int i = 1; i < 8; ++i) mm = fmaxf(mm, sRed[i]);
    sM = mm;
  }
  __syncthreads();
  float s = 0.0f;
  for (int i = tid; i < VV; i += blockDim.x) s += expf(row[i] - sM);
  s = waveSum(s);
  if (lane == 0) sRed[wid] = s;
  __syncthreads();
  if (tid == 0) {
    float ss = 0.0f;
    for (int i = 0; i < 8; ++i) ss += sRed[i];
    lse[b] = sM + logf(ss);
  }
}

__global__ void k_finalize(float* __restrict__ out, const float* __restrict__ lse) {
  size_t i = (size_t)blockIdx.x * blockDim.x + threadIdx.x;
  if (i >= (size_t)BB * VV) return;
  out[i] -= lse[i / VV];
}

// ---------------- launcher ----------------
extern "C" void kernel_launch(void* const* d_in, const int* in_sizes, int n_in,
                              void* d_out, int out_size, void* d_ws, size_t ws_size,
                              hipStream_t stream) {
  (void)in_sizes; (void)n_in; (void)out_size; (void)ws_size;
  const int*   x        = (const int*)  d_in[0];
  const float* enc      = (const float*)d_in[1];
  const float* hidden   = (const float*)d_in[2];
  const float* embed    = (const float*)d_in[3];
  const float* W_ih_l0  = (const float*)d_in[4];
  const float* W_hh_l0  = (const float*)d_in[5];
  const float* b_ih_l0  = (const float*)d_in[6];
  const float* b_hh_l0  = (const float*)d_in[7];
  const float* W_ih_l0r = (const float*)d_in[8];
  const float* W_hh_l0r = (const float*)d_in[9];
  const float* b_ih_l0r = (const float*)d_in[10];
  const float* b_hh_l0r = (const float*)d_in[11];
  const float* W_ih_l1  = (const float*)d_in[12];
  const float* W_hh_l1  = (const float*)d_in[13];
  const float* b_ih_l1  = (const float*)d_in[14];
  const float* b_hh_l1  = (const float*)d_in[15];
  const float* W_ih_l1r = (const float*)d_in[16];
  const float* W_hh_l1r = (const float*)d_in[17];
  const float* b_ih_l1r = (const float*)d_in[18];
  const float* b_hh_l1r = (const float*)d_in[19];
  const float* W_fnn    = (const float*)d_in[20];
  // d_in[21] = b_fnn: softmax-shift invariant -> unused
  const float* W_cat    = (const float*)d_in[22];
  const float* b_cat    = (const float*)d_in[23];
  const float* W_out    = (const float*)d_in[24];
  const float* b_out    = (const float*)d_in[25];
  float* out = (float*)d_out;

  float* ws   = (float*)d_ws;
  float* e    = ws;                   // 64*100
  float* gi   = e    + BB * EE;       // 64*1500
  float* gh   = gi   + BB * 3 * HD;   // 64*1500
  float* gi2  = gh   + BB * 3 * HD;   // 64*1500
  float* gh2  = gi2  + BB * 3 * HD;   // 64*1500
  float* l0   = gh2  + BB * 3 * HD;   // 64*1000
  float* hcat = l0   + BB * HH;       // 64*2000  (h | context)
  float* hW   = hcat + BB * 2 * HH;   // 64*1000
  float* hc   = hW   + BB * HH;       // 64*1000
  float* lse  = hc   + BB * HH;       // 64

  const dim3 blk(256);
  auto ggrid = [](int N) { int t = (N + 15) / 16; return dim3((t + 7) / 8); };

  k_embed<<<dim3((BB * EE + 255) / 256), blk, 0, stream>>>(x, embed, e);

  // ---- GRU layer 0 (fwd + bwd) ----
  k_gemm64<true><<<ggrid(3 * HD), blk, 0, stream>>>(e, EE, W_ih_l0, b_ih_l0, gi, 3 * HD, 3 * HD, EE, 0);
  k_gemm64<true><<<ggrid(3 * HD), blk, 0, stream>>>(hidden + 0 * BB * HD, HD, W_hh_l0, b_hh_l0, gh, 3 * HD, 3 * HD, HD, 0);
  k_gemm64<true><<<ggrid(3 * HD), blk, 0, stream>>>(e, EE, W_ih_l0r, b_ih_l0r, gi2, 3 * HD, 3 * HD, EE, 0);
  k_gemm64<true><<<ggrid(3 * HD), blk, 0, stream>>>(hidden + 1 * BB * HD, HD, W_hh_l0r, b_hh_l0r, gh2, 3 * HD, 3 * HD, HD, 0);
  k_gate<<<dim3((BB * HD + 255) / 256), blk, 0, stream>>>(gi,  gh,  hidden + 0 * BB * HD, l0, 0,  HH);
  k_gate<<<dim3((BB * HD + 255) / 256), blk, 0, stream>>>(gi2, gh2, hidden + 1 * BB * HD, l0, HD, HH);

  // ---- GRU layer 1 (fwd + bwd) -> hcat[:, 0:1000] ----
  k_gemm64<true><<<ggrid(3 * HD), blk, 0, stream>>>(l0, HH, W_ih_l1, b_ih_l1, gi, 3 * HD, 3 * HD, HH, 0);
  k_gemm64<true><<<ggrid(3 * HD), blk, 0, stream>>>(hidden + 2 * BB * HD, HD, W_hh_l1, b_hh_l1, gh, 3 * HD, 3 * HD, HD, 0);
  k_gemm64<true><<<ggrid(3 * HD), blk, 0, stream>>>(l0, HH, W_ih_l1r, b_ih_l1r, gi2, 3 * HD, 3 * HD, HH, 0);
  k_gemm64<true><<<ggrid(3 * HD), blk, 0, stream>>>(hidden + 3 * BB * HD, HD, W_hh_l1r, b_hh_l1r, gh2, 3 * HD, 3 * HD, HD, 0);
  k_gate<<<dim3((BB * HD + 255) / 256), blk, 0, stream>>>(gi,  gh,  hidden + 2 * BB * HD, hcat, 0,  2 * HH);
  k_gate<<<dim3((BB * HD + 255) / 256), blk, 0, stream>>>(gi2, gh2, hidden + 3 * BB * HD, hcat, HD, 2 * HH);

  // ---- attention (algebraically collapsed): hW = h @ W_fnn ----
  k_gemm64<false><<<ggrid(HH), blk, 0, stream>>>(hcat, 2 * HH, W_fnn, (const float*)nullptr, hW, HH, HH, HH, 0);
  k_attention<<<dim3(BB), blk, 0, stream>>>(hcat, hW, enc);

  // ---- hc = tanh([h|c] @ W_cat^T + b_cat) ----
  k_gemm64<true><<<ggrid(HH), blk, 0, stream>>>(hcat, 2 * HH, W_cat, b_cat, hc, HH, HH, 2 * HH, 1);

  // ---- logits = hc @ W_out^T + b_out (memory-bound: 128 MB of W_out) ----
  k_gemm64<true><<<ggrid(VV), blk, 0, stream>>>(hc, HH, W_out, b_out, out, VV, VV, HH, 0);

  // ---- log_softmax ----
  k_lse<<<dim3(BB), blk, 0, stream>>>(out, lse);
  k_finalize<<<dim3((int)(((size_t)BB * VV + 255) / 256)), blk, 0, stream>>>(out, lse);
}